// BPBookLayer_63410897158471
// MI455X (gfx1250) — compile-verified
//
#include <hip/hip_runtime.h>
#include <math.h>

// ---------------- problem constants ----------------
#define B_SZ   32
#define L_SZ   2048
#define D_SZ   1024
#define P_SZ   1024
#define TOPK   5
#define ALPHA  0.1f
#define EPS    1e-12f

typedef __attribute__((ext_vector_type(2))) float v2f;
typedef __attribute__((ext_vector_type(8))) float v8f;

// ---------------- workspace layout (floats) ----------------
// qsum   : [32,1024]  raw column sums of x over L            off 0
// invq   : [32]       1/max(||qsum_row||, eps)               off 32768
// invp   : [1024]     1/max(||proto_row||, eps)              off 32800
// scores : [32,1024]  cosine scores                          off 33824
// attn   : [32,5]     softmax weights                        off 66592
// tidx   : [32,5] int top-k indices                          off 66752
// agg    : [32,1024]  weighted prototype aggregate           off 66912
#define WS_QSUM   0
#define WS_INVQ   32768
#define WS_INVP   32800
#define WS_SCORES 33824
#define WS_ATTN   66592
#define WS_TIDX   66752
#define WS_AGG    66912

// ---------------- kernel 0: zero qsum ----------------
__global__ void k_zero(float* __restrict__ buf, int n) {
  int i = blockIdx.x * blockDim.x + threadIdx.x;
  if (i < n) buf[i] = 0.0f;
}

// ---------------- kernel 1: mean pool partial sums (atomic) ----------------
// grid (4 dchunk, 16 lchunk, 32 b), block 256. Coalesced over d.
__global__ void k_pool(const float* __restrict__ x, float* __restrict__ qsum) {
  int d = blockIdx.x * 256 + threadIdx.x;
  int b = blockIdx.z;
  int l0 = blockIdx.y * (L_SZ / 16);
  const float* xp = x + (size_t)b * L_SZ * D_SZ + (size_t)l0 * D_SZ + d;
  float s = 0.0f;
  #pragma unroll 4
  for (int l = 0; l < L_SZ / 16; ++l) s += xp[(size_t)l * D_SZ];
  atomicAdd(&qsum[b * D_SZ + d], s);
}

// ---------------- kernel 2a: query inverse norms ----------------
// 1 block, 32 threads; thread b handles one query row.
__global__ void k_qnorm(const float* __restrict__ qsum, float* __restrict__ invq) {
  int b = threadIdx.x;
  const float* q = qsum + b * D_SZ;
  float s = 0.0f;
  for (int d = 0; d < D_SZ; ++d) { float v = q[d]; s += v * v; }
  invq[b] = 1.0f / fmaxf(sqrtf(s), EPS);
}

// ---------------- kernel 2b: prototype inverse norms ----------------
// one wave (32 threads) per row; coalesced lane-strided read + shuffle reduce.
__global__ void k_pnorm(const float* __restrict__ proto, float* __restrict__ invp) {
  int p = blockIdx.x;
  int lane = threadIdx.x;
  const float* row = proto + (size_t)p * D_SZ;
  float s = 0.0f;
  for (int d = lane; d < D_SZ; d += 32) { float v = row[d]; s += v * v; }
  #pragma unroll
  for (int off = 16; off > 0; off >>= 1)
    s += __shfl_down(s, off, 32);
  if (lane == 0) invp[p] = 1.0f / fmaxf(sqrtf(s), EPS);
}

// ---------------- kernel 3: cosine score GEMM via V_WMMA_F32_16X16X4_F32 ----
// grid (64 ntiles, 2 mtiles), block 32 (one wave, EXEC all ones).
// S[m,n] = (qsum[m,:] . proto[n,:]) * invq[m] * invp[n]
// A fragment (16x4 f32): lane m=l&15; VGPR0/1 hold K = k+2*(l>>4) {+0,+1}.
// B fragment (4x16 f32): lane n=l&15; same K pairing (B row striped per VGPR).
// C/D (16x16 f32): VGPR r -> M = r + 8*(l>>4), N = l&15.
__global__ void k_scores(const float* __restrict__ qsum,
                         const float* __restrict__ proto,
                         const float* __restrict__ invq,
                         const float* __restrict__ invp,
                         float* __restrict__ scores) {
  int lane = threadIdx.x;
  int nt   = blockIdx.x;       // ntile: p columns [nt*16, nt*16+16)
  int mt   = blockIdx.y;       // mtile: b rows    [mt*16, mt*16+16)
  int m    = lane & 15;
  int hi   = lane >> 4;

  const float* arow = qsum  + (size_t)(mt * 16 + m) * D_SZ;   // A row (M = lane&15)
  const float* brow = proto + (size_t)(nt * 16 + m) * D_SZ;   // B col => proto row (N = lane&15)

  v8f c = {};
  for (int k = 0; k < D_SZ; k += 4) {
    int kk = k + 2 * hi;
    v2f a = *(const v2f*)(arow + kk);   // K=kk, kk+1
    v2f b = *(const v2f*)(brow + kk);   // K=kk, kk+1
    c = __builtin_amdgcn_wmma_f32_16x16x4_f32(
        /*neg_a=*/false, a, /*neg_b=*/false, b,
        /*c_mod=*/(short)0, c, /*reuse_a=*/false, /*reuse_b=*/false);
  }

  #pragma unroll
  for (int r = 0; r < 8; ++r) {
    int row = mt * 16 + r + 8 * hi;     // batch index
    int col = nt * 16 + m;              // prototype index
    scores[row * P_SZ + col] = c[r] * invq[row] * invp[col];
  }
}

// ---------------- kernel 4: top-5 + softmax ----------------
// 1 block, 32 threads; thread b scans its 1024-score row serially.
// Strict '>' insertion reproduces jax.lax.top_k tie-breaking (earliest index).
__global__ void k_topk(const float* __restrict__ scores,
                       float* __restrict__ attn, int* __restrict__ tidx) {
  int b = threadIdx.x;
  const float* row = scores + b * P_SZ;
  float val[TOPK];
  int   idx[TOPK];
  #pragma unroll
  for (int k = 0; k < TOPK; ++k) { val[k] = -INFINITY; idx[k] = 0; }
  for (int p = 0; p < P_SZ; ++p) {
    float v = row[p];
    if (v > val[TOPK - 1]) {
      int j = TOPK - 1;
      while (j > 0 && v > val[j - 1]) { val[j] = val[j - 1]; idx[j] = idx[j - 1]; --j; }
      val[j] = v; idx[j] = p;
    }
  }
  float mx = val[0];
  float e[TOPK], sum = 0.0f;
  #pragma unroll
  for (int k = 0; k < TOPK; ++k) { e[k] = __expf(val[k] - mx); sum += e[k]; }
  float inv = 1.0f / sum;
  #pragma unroll
  for (int k = 0; k < TOPK; ++k) {
    attn[b * TOPK + k] = e[k] * inv;
    tidx[b * TOPK + k] = idx[k];
  }
}

// ---------------- kernel 5: weighted prototype aggregate ----------------
// grid (4 dchunk, 32 b), block 256.
__global__ void k_agg(const float* __restrict__ proto,
                      const float* __restrict__ attn,
                      const int* __restrict__ tidx,
                      float* __restrict__ agg) {
  int d = blockIdx.x * 256 + threadIdx.x;
  int b = blockIdx.y;
  float s = 0.0f;
  #pragma unroll
  for (int k = 0; k < TOPK; ++k)
    s += attn[b * TOPK + k] * proto[(size_t)tidx[b * TOPK + k] * D_SZ + d];
  agg[b * D_SZ + d] = s;
}

// ---------------- kernel 6: broadcast residual add (float4 streamed) -------
__global__ void k_resadd(const float* __restrict__ x,
                         const float* __restrict__ agg,
                         float* __restrict__ out, int n4) {
  int i = blockIdx.x * blockDim.x + threadIdx.x;
  int stride = gridDim.x * blockDim.x;
  for (; i < n4; i += stride) {
    long e = (long)i * 4;
    int d = (int)(e & (D_SZ - 1));
    int b = (int)(e >> 21);            // 2048*1024 = 2^21 elems per batch
    float4 xv = ((const float4*)x)[i];
    float4 av = *(const float4*)(agg + b * D_SZ + d);
    float4 o;
    o.x = fmaf(ALPHA, av.x, xv.x);
    o.y = fmaf(ALPHA, av.y, xv.y);
    o.z = fmaf(ALPHA, av.z, xv.z);
    o.w = fmaf(ALPHA, av.w, xv.w);
    ((float4*)out)[i] = o;
  }
}

// ---------------- launcher ----------------
extern "C" void kernel_launch(void* const* d_in, const int* in_sizes, int n_in,
                              void* d_out, int out_size, void* d_ws, size_t ws_size,
                              hipStream_t stream) {
  const float* x     = (const float*)d_in[0];   // [32,2048,1024]
  const float* proto = (const float*)d_in[1];   // [1024,1024]
  float* out = (float*)d_out;
  float* ws  = (float*)d_ws;

  float* qsum   = ws + WS_QSUM;
  float* invq   = ws + WS_INVQ;
  float* invp   = ws + WS_INVP;
  float* scores = ws + WS_SCORES;
  float* attn   = ws + WS_ATTN;
  int*   tidx   = (int*)(ws + WS_TIDX);
  float* agg    = ws + WS_AGG;

  // 0) zero the atomic accumulator
  k_zero<<<(B_SZ * D_SZ) / 256, 256, 0, stream>>>(qsum, B_SZ * D_SZ);

  // 1) mean pool (raw sums; scale cancels in cosine)
  k_pool<<<dim3(D_SZ / 256, 16, B_SZ), 256, 0, stream>>>(x, qsum);

  // 2) inverse norms
  k_qnorm<<<1, 32, 0, stream>>>(qsum, invq);
  k_pnorm<<<P_SZ, 32, 0, stream>>>(proto, invp);

  // 3) WMMA fp32 cosine score GEMM: [32,1024] x [1024,1024]^T
  k_scores<<<dim3(P_SZ / 16, B_SZ / 16), 32, 0, stream>>>(qsum, proto, invq, invp, scores);

  // 4) top-5 + softmax
  k_topk<<<1, 32, 0, stream>>>(scores, attn, tidx);

  // 5) weighted gather-aggregate of raw prototypes
  k_agg<<<dim3(D_SZ / 256, B_SZ), 256, 0, stream>>>(proto, attn, tidx, agg);

  // 6) residual broadcast add (memory-bound streaming pass)
  int n4 = (B_SZ * L_SZ * D_SZ) / 4;
  k_resadd<<<16384, 256, 0, stream>>>(x, agg, out, n4);
}